// OrthogonalLayer1D_10170482557088
// MI455X (gfx1250) — compile-verified
//
#include <hip/hip_runtime.h>

// Gram–Schmidt over 8 vectors of dim 1024, batched 4096x (fp32).
// Strategy (memory-bound: 256 MB traffic -> ~11us @ 23.3 TB/s):
//   one block per batch row; X tile (8x1024, 32KB) staged once in LDS;
//   G = X X^T via v_wmma_f32_16x16x4_f32 (8 waves split K, LDS-reduce);
//   tiny sequential recurrence on G -> coefficient matrix A (thread 0);
//   Q = A * X on VALU with coalesced float4 stores.

typedef __attribute__((ext_vector_type(2))) float v2f;
typedef __attribute__((ext_vector_type(8))) float v8f;

#define NM    8      // number of model vectors
#define NC    1024   // vector length
#define ROWP  1028   // padded LDS row stride (floats): 1028 % 64 == 4 -> no bank conflicts
#define EPSF  1e-8f

__global__ __launch_bounds__(256)
void gs_wmma_kernel(const float* __restrict__ x, float* __restrict__ out, int nb) {
  __shared__ float Xs[NM * ROWP];       // 8 x 1028 fp32  (~32.9 KB)
  __shared__ float Gpart[8][8][8];      // [wave][i][j] partial Gram
  __shared__ float Gm[8][8];            // reduced Gram
  __shared__ float Ac[8][8];            // GS coefficient matrix (q_i = sum_k Ac[i][k] x_k)

  const int b    = blockIdx.x;
  const int t    = threadIdx.x;
  const int lane = t & 31;
  const int wave = t >> 5;

  // ---------------- Stage 0: global -> LDS, one float4 per row per thread ----
  {
    const float4* __restrict__ xv = (const float4*)x;
    #pragma unroll
    for (int row = 0; row < NM; ++row) {
      float4 v = xv[((size_t)row * nb + b) * (NC / 4) + t];
      *(float4*)&Xs[row * ROWP + t * 4] = v;
    }
  }
  __syncthreads();

  // ---------------- Stage 1: Gram matrix via WMMA f32 16x16x4 ----------------
  // D[m][n] += sum_k A[m][k] * B[k][n], with A = X-slice (rows 8..15 zero)
  // and B = A^T. ISA layout makes A and B per-lane register images identical:
  //   A: lane = m + 16*(k>=2), vgpr = k&1 ; B: lane = n + 16*(k>=2), vgpr = k&1
  {
    const int   m       = lane & 15;
    const int   kb      = (lane >> 4) << 1;      // 0 or 2
    const int   rowSafe = m & 7;                 // always a valid LDS row
    const float keep    = (m < NM) ? 1.0f : 0.0f;
    const int   kbase   = wave * 128 + kb;       // each wave covers K-range of 128

    v8f acc = {};
    for (int k0 = 0; k0 < 128; k0 += 4) {
      v2f a = *(const v2f*)&Xs[rowSafe * ROWP + kbase + k0];
      a.x *= keep;
      a.y *= keep;
      acc = __builtin_amdgcn_wmma_f32_16x16x4_f32(
          /*neg_a=*/false, a, /*neg_b=*/false, a,
          /*c_mod=*/(short)0, acc, /*reuse_a=*/false, /*reuse_b=*/false);
    }
    // C/D layout: VGPR r holds (M=r, N=lane) for lanes 0..15. Keep top-left 8x8.
    if (lane < 8) {
      #pragma unroll
      for (int r = 0; r < 8; ++r) Gpart[wave][r][lane] = acc[r];
    }
  }
  __syncthreads();

  // Reduce the 8 per-wave partial Gram matrices.
  if (t < 64) {
    const int i = t >> 3, j = t & 7;
    float s = 0.0f;
    #pragma unroll
    for (int w = 0; w < 8; ++w) s += Gpart[w][i][j];
    Gm[i][j] = s;
  }
  __syncthreads();

  // ---------------- Stage 2: sequential GS recurrence on G (thread 0) --------
  // Tracks q_i = sum_k A[i][k] x_k. Projections use ORIGINAL x_i (as in the
  // reference), so proj_ij = (sum_k A[j][k] G[k][i]) / (d_j + EPS).
  if (t == 0) {
    float g[8][8];
    #pragma unroll
    for (int i = 0; i < 8; ++i)
      #pragma unroll
      for (int j = 0; j < 8; ++j) g[i][j] = Gm[i][j];

    float A[8][8];
    #pragma unroll
    for (int i = 0; i < 8; ++i)
      #pragma unroll
      for (int j = 0; j < 8; ++j) A[i][j] = 0.0f;

    float d[8];

    // i = 0: q0 = x0 / (||x0|| + EPS)
    {
      float n0   = sqrtf(fmaxf(g[0][0], 0.0f));
      float inv0 = 1.0f / (n0 + EPSF);
      A[0][0] = inv0;
      d[0]    = inv0 * inv0 * g[0][0];
    }

    #pragma unroll
    for (int i = 1; i < 8; ++i) {
      float c[8];
      #pragma unroll
      for (int k = 0; k < 8; ++k) c[k] = 0.0f;
      c[i] = 1.0f;

      #pragma unroll
      for (int j = 0; j < 8; ++j) {
        if (j < i) {
          float dv = 0.0f;
          #pragma unroll
          for (int k = 0; k < 8; ++k)
            if (k <= j) dv += A[j][k] * g[k][i];
          float proj = dv / (d[j] + EPSF);
          #pragma unroll
          for (int k = 0; k < 8; ++k)
            if (k <= j) c[k] -= proj * A[j][k];
        }
      }

      // norm^2 = c^T G c
      float n2 = 0.0f;
      #pragma unroll
      for (int k = 0; k < 8; ++k) {
        if (k <= i) {
          float y = 0.0f;
          #pragma unroll
          for (int m2 = 0; m2 < 8; ++m2)
            if (m2 <= i) y += g[k][m2] * c[m2];
          n2 += c[k] * y;
        }
      }
      float nn = sqrtf(fmaxf(n2, 0.0f));
      if (nn > EPSF) {
        float inv = 1.0f / nn;
        #pragma unroll
        for (int k = 0; k < 8; ++k) c[k] *= inv;
        d[i] = n2 * inv * inv;
      } else {
        d[i] = n2;   // reference keeps q_i unnormalized when norm <= EPS
      }
      #pragma unroll
      for (int k = 0; k < 8; ++k) A[i][k] = c[k];
    }

    #pragma unroll
    for (int i = 0; i < 8; ++i)
      #pragma unroll
      for (int k = 0; k < 8; ++k) Ac[i][k] = A[i][k];
  }
  __syncthreads();

  // ---------------- Stage 3: Q = Ac * X, coalesced float4 stores --------------
  {
    float4 xk[8];
    #pragma unroll
    for (int k = 0; k < 8; ++k) xk[k] = *(const float4*)&Xs[k * ROWP + t * 4];

    float4* __restrict__ ov = (float4*)out;
    #pragma unroll
    for (int m = 0; m < 8; ++m) {
      float4 acc4 = {0.0f, 0.0f, 0.0f, 0.0f};
      #pragma unroll
      for (int k = 0; k < 8; ++k) {
        const float c = Ac[m][k];   // LDS broadcast
        acc4.x += c * xk[k].x;
        acc4.y += c * xk[k].y;
        acc4.z += c * xk[k].z;
        acc4.w += c * xk[k].w;
      }
      ov[((size_t)m * nb + b) * (NC / 4) + t] = acc4;
    }
  }
}

extern "C" void kernel_launch(void* const* d_in, const int* in_sizes, int n_in,
                              void* d_out, int out_size, void* d_ws, size_t ws_size,
                              hipStream_t stream) {
  const float* x   = (const float*)d_in[0];
  float*       out = (float*)d_out;
  const int    nb  = in_sizes[0] / (NM * NC);   // 4096 for the reference shapes
  gs_wmma_kernel<<<nb, 256, 0, stream>>>(x, out, nb);
}